// HierKalmanVAE_V2_38139309588725
// MI455X (gfx1250) — compile-verified
//
#include <hip/hip_runtime.h>

// HierKalmanVAE forward for MI455X (gfx1250, wave32).
// One wave (32 lanes) per batch element; T=96 sequential Kalman steps with
// mu/Sigma resident in LDS; all matmuls via V_WMMA_F32_16X16X4_F32.

constexpr int TT = 96;
constexpr int BB = 128;
constexpr int ZZ = 32;   // latent dim
constexpr int AD = 16;   // obs dim

typedef __attribute__((ext_vector_type(2))) float v2f;
typedef __attribute__((ext_vector_type(8))) float v8f;

// ---- WMMA f32 16x16x4 tile helpers (layouts per CDNA5 ISA 7.12.2) ----
// A 16x4: lane l -> m = l&15, k = 2*(l>>4) + {0,1}
template <bool TRANS>
__device__ __forceinline__ v2f ldAtile(const float* M, int ld, int r0, int c0,
                                       int lane, float alpha) {
  int m  = lane & 15;
  int kb = (lane >> 4) << 1;
  v2f a;
  if (TRANS) {  // logical A(m,k) = M[k][m]
    a.x = M[(c0 + kb    ) * ld + (r0 + m)];
    a.y = M[(c0 + kb + 1) * ld + (r0 + m)];
  } else {
    a.x = M[(r0 + m) * ld + (c0 + kb    )];
    a.y = M[(r0 + m) * ld + (c0 + kb + 1)];
  }
  a.x *= alpha; a.y *= alpha;
  return a;
}

// B 4x16: lane l -> n = l&15, k = 2*(l>>4) + {0,1}
template <bool TRANS>
__device__ __forceinline__ v2f ldBtile(const float* M, int ld, int r0, int c0,
                                       int lane) {
  int n  = lane & 15;
  int kb = (lane >> 4) << 1;
  v2f b;
  if (TRANS) {  // logical B(k,n) = M[n][k]
    b.x = M[(c0 + n) * ld + (r0 + kb    )];
    b.y = M[(c0 + n) * ld + (r0 + kb + 1)];
  } else {
    b.x = M[(r0 + kb    ) * ld + (c0 + n)];
    b.y = M[(r0 + kb + 1) * ld + (c0 + n)];
  }
  return b;
}

// C/D 16x16: VGPR v, lane l -> row = v + 8*(l>>4), col = l&15
__device__ __forceinline__ v8f ldCtile(const float* M, int ld, int r0, int c0,
                                       int lane) {
  int n = lane & 15, hi = lane >> 4;
  v8f c;
#pragma unroll
  for (int v = 0; v < 8; ++v) c[v] = M[(r0 + v + 8 * hi) * ld + (c0 + n)];
  return c;
}

__device__ __forceinline__ void stCtile(float* M, int ld, int r0, int c0,
                                        int lane, v8f c) {
  int n = lane & 15, hi = lane >> 4;
#pragma unroll
  for (int v = 0; v < 8; ++v) M[(r0 + v + 8 * hi) * ld + (c0 + n)] = c[v];
}

// D(MxN) (+)= alpha * op(A)(MxK) * op(B)(KxN), fully unrolled WMMA tiles.
template <int M, int N, int K, bool TA, bool TB, bool ACC>
__device__ __forceinline__ void wmma_mm(float alpha, const float* A, int lda,
                                        const float* B, int ldb, float* Dm,
                                        int ldd) {
  const int lane = threadIdx.x;
#pragma unroll
  for (int mt = 0; mt < M; mt += 16) {
#pragma unroll
    for (int nt = 0; nt < N; nt += 16) {
      v8f c = {};
      if (ACC) c = ldCtile(Dm, ldd, mt, nt, lane);
#pragma unroll
      for (int kt = 0; kt < K; kt += 4) {
        v2f a = ldAtile<TA>(A, lda, mt, kt, lane, alpha);
        v2f b = ldBtile<TB>(B, ldb, kt, nt, lane);
        c = __builtin_amdgcn_wmma_f32_16x16x4_f32(false, a, false, b,
                                                  (short)0, c, false, false);
      }
      stCtile(Dm, ldd, mt, nt, lane, c);
    }
  }
}

// ---- data-independent latent outputs (pure bandwidth) ----
__global__ void fill_latents(float* __restrict__ lm, float* __restrict__ lv) {
  const int stride = gridDim.x * blockDim.x;
  int idx = blockIdx.x * blockDim.x + threadIdx.x;
  const int NV = TT * BB * 2 * ZZ * ZZ;
  for (int i = idx; i < NV; i += stride) {
    int j  = i & 31;
    int r  = (i >> 5) & 31;
    int l  = (i >> 10) & 1;
    int t  = (i >> 11) / BB;
    float v = 0.0f;
    if (r == j) {
      if (l == 0) v = (t >= 2) ? 0.08f : 0.0f;   // O from t>=FACTOR
      else        v = (t < 4) ? 20.0f : 0.0f;    // sigma0 until first jump
    }
    lv[i] = v;
  }
  const int NM = TT * BB * ZZ;  // level-0 means are zero
  for (int i = idx; i < NM; i += stride) {
    int z  = i & 31;
    int tb = i >> 5;
    lm[(tb * 2) * ZZ + z] = 0.0f;
  }
}

// ---- main sequential Kalman kernel: 1 wave per batch element ----
__global__ __launch_bounds__(32) void kalman_fwd(
    const float* __restrict__ obsg, const float* __restrict__ Ag,
    const float* __restrict__ Cg, const float* __restrict__ Dg,
    float* __restrict__ mu_filt, float* __restrict__ sigma_filt,
    float* __restrict__ mu_pred, float* __restrict__ sigma_pred,
    float* __restrict__ lat_means, float* __restrict__ S_out) {
  __shared__ float sSigma[ZZ * 33];
  __shared__ float sSigmaZ[ZZ * 33];
  __shared__ float sTmp[ZZ * 33];
  __shared__ float sImat[ZZ * 33];
  __shared__ float sAn[ZZ * 33];
  __shared__ float sDn[ZZ * 33];
  __shared__ float sCt[AD * 33];
  __shared__ float sSaug[AD * 33];  // [16 x 32] augmented for Gauss-Jordan
  __shared__ float sU[ZZ * 17];
  __shared__ float sK[ZZ * 17];
  __shared__ float sMu[ZZ];
  __shared__ float sMuZ[ZZ];
  __shared__ float sMuJ[ZZ];
  __shared__ float sRv[AD];

  const int lane = threadIdx.x;
  const int b = blockIdx.x;

  // state init: mu0 = 0, Sigma0 = 20 I, chain mu_j = 0.01
  sMu[lane] = 0.0f;
  sMuJ[lane] = 0.01f;
  for (int i = lane; i < ZZ * ZZ; i += 32) {
    int r = i >> 5, c = i & 31;
    sSigma[r * 33 + c] = (r == c) ? 20.0f : 0.0f;
  }
  __syncthreads();

  for (int t = 0; t < TT; ++t) {
    const int tb = t * BB + b;

    // prefetch next step's A/C/D (global_prefetch_b8)
    if (t + 1 < TT) {
      __builtin_prefetch(Cg + ((size_t)b * TT + (t + 1)) * AD * ZZ + lane * 16, 0, 1);
      __builtin_prefetch(Ag + (((size_t)b * TT + (t + 1)) * 3) * ZZ * ZZ + lane * 32, 0, 1);
      __builtin_prefetch(Dg + (((size_t)b * TT + (t + 1)) * 2) * ZZ * ZZ + lane * 32, 0, 1);
    }

    // emit priors (mu_pred, sigma_pred)
    mu_pred[tb * ZZ + lane] = sMu[lane];
    for (int i = lane; i < ZZ * ZZ; i += 32)
      sigma_pred[(size_t)tb * ZZ * ZZ + i] = sSigma[(i >> 5) * 33 + (i & 31)];

    // stage C_t [16 x 32]
    const float* Cp = Cg + ((size_t)b * TT + t) * AD * ZZ;
    for (int i = lane; i < AD * ZZ; i += 32)
      sCt[(i >> 5) * 33 + (i & 31)] = Cp[i];

    // level-1 chain: jump every f2 = 4 steps via A[:,t,2]
    if (t > 0 && (t & 3) == 0) {
      const float* A2 = Ag + (((size_t)b * TT + t) * 3 + 2) * ZZ * ZZ;
      float acc = 0.0f;
      for (int k = 0; k < ZZ; ++k) acc += A2[lane * ZZ + k] * sMuJ[k];
      __syncthreads();
      sMuJ[lane] = acc;
    }
    lat_means[((size_t)tb * 2 + 1) * ZZ + lane] = sMuJ[lane];
    __syncthreads();

    // innovation r = o - C mu
    if (lane < AD) {
      float acc = obsg[tb * AD + lane];
      for (int k = 0; k < ZZ; ++k) acc -= sCt[lane * 33 + k] * sMu[k];
      sRv[lane] = acc;
    }
    __syncthreads();

    // S = C Sigma C^T + R  (into left half of augmented matrix)
    wmma_mm<16, 32, 32, false, false, false>(1.0f, sCt, 33, sSigma, 33, sTmp, 33);
    wmma_mm<16, 16, 32, false, true, false>(1.0f, sTmp, 33, sCt, 33, sSaug, 33);
    for (int i = lane; i < AD * AD; i += 32) {
      int r = i >> 4, c = i & 15;
      float s = sSaug[r * 33 + c] + ((r == c) ? 0.03f : 0.0f);
      sSaug[r * 33 + c] = s;
      S_out[(size_t)tb * AD * AD + i] = s;
      sSaug[r * 33 + 16 + c] = (r == c) ? 1.0f : 0.0f;  // augmented identity
    }
    __syncthreads();

    // invert S: Gauss-Jordan, each lane owns one augmented column
    for (int i = 0; i < AD; ++i) {
      float inv = 1.0f / sSaug[i * 33 + i];
      float rowi = sSaug[i * 33 + lane] * inv;
      sSaug[i * 33 + lane] = rowi;
      for (int r = 0; r < AD; ++r) {
        if (r == i) continue;
        float f = sSaug[r * 33 + i];
        sSaug[r * 33 + lane] -= f * rowi;
      }
    }
    __syncthreads();

    // U = Sigma C^T (32x16);  K = U * Sinv (32x16)
    wmma_mm<32, 16, 32, false, true, false>(1.0f, sSigma, 33, sCt, 33, sU, 17);
    wmma_mm<32, 16, 16, false, false, false>(1.0f, sU, 17, sSaug + 16, 33, sK, 17);

    // mu_z = mu + K r
    {
      float acc = sMu[lane];
      for (int k = 0; k < AD; ++k) acc += sK[lane * 17 + k] * sRv[k];
      sMuZ[lane] = acc;
      mu_filt[tb * ZZ + lane] = acc;
    }
    __syncthreads();

    // Imat = I - K C
    wmma_mm<32, 32, 16, false, false, false>(1.0f, sK, 17, sCt, 33, sImat, 33);
    for (int i = lane; i < ZZ * ZZ; i += 32) {
      int r = i >> 5, c = i & 31;
      sImat[r * 33 + c] = ((r == c) ? 1.0f : 0.0f) - sImat[r * 33 + c];
    }
    __syncthreads();

    // sigma_z = Imat Sigma Imat^T + 0.03 K K^T   (Joseph form)
    wmma_mm<32, 32, 32, false, false, false>(1.0f, sImat, 33, sSigma, 33, sTmp, 33);
    wmma_mm<32, 32, 32, false, true, false>(1.0f, sTmp, 33, sImat, 33, sSigmaZ, 33);
    wmma_mm<32, 32, 16, false, true, true>(0.03f, sK, 17, sK, 17, sSigmaZ, 33);
    for (int i = lane; i < ZZ * ZZ; i += 32)
      sigma_filt[(size_t)tb * ZZ * ZZ + i] = sSigmaZ[(i >> 5) * 33 + (i & 31)];
    __syncthreads();

    // predict with A[:,t+1,0], D[:,t+1,0] (last step's predict is unused)
    if (t + 1 < TT) {
      const float* Ap = Ag + (((size_t)b * TT + (t + 1)) * 3 + 0) * ZZ * ZZ;
      const float* Dp = Dg + (((size_t)b * TT + (t + 1)) * 2 + 0) * ZZ * ZZ;
      for (int i = lane; i < ZZ * ZZ; i += 32) {
        sAn[(i >> 5) * 33 + (i & 31)] = Ap[i];
        sDn[(i >> 5) * 33 + (i & 31)] = Dp[i];
      }
      __syncthreads();

      // mu' = A mu_z + D m1
      float acc = 0.0f;
      for (int k = 0; k < ZZ; ++k)
        acc += sAn[lane * 33 + k] * sMuZ[k] + sDn[lane * 33 + k] * sMuJ[k];
      sMu[lane] = acc;

      // Sigma' = A sigma_z A^T + Q (+ 20 D D^T while var1 = sigma0)
      wmma_mm<32, 32, 32, false, false, false>(1.0f, sAn, 33, sSigmaZ, 33, sTmp, 33);
      wmma_mm<32, 32, 32, false, true, false>(1.0f, sTmp, 33, sAn, 33, sSigma, 33);
      if (t < 4)
        wmma_mm<32, 32, 32, false, true, true>(20.0f, sDn, 33, sDn, 33, sSigma, 33);
      sSigma[lane * 33 + lane] += 0.08f;  // Q
      __syncthreads();
    }
  }
}

extern "C" void kernel_launch(void* const* d_in, const int* in_sizes, int n_in,
                              void* d_out, int out_size, void* d_ws,
                              size_t ws_size, hipStream_t stream) {
  const float* obs = (const float*)d_in[0];
  const float* A = (const float*)d_in[1];
  const float* C = (const float*)d_in[2];
  const float* D = (const float*)d_in[3];
  float* out = (float*)d_out;

  const size_t n_mu = (size_t)TT * BB * ZZ;            // 393216
  const size_t n_sig = (size_t)TT * BB * ZZ * ZZ;      // 12582912
  const size_t n_lm = (size_t)TT * BB * 2 * ZZ;        // 786432
  const size_t n_lv = (size_t)TT * BB * 2 * ZZ * ZZ;   // 25165824

  float* mu_filt = out;
  float* sigma_filt = mu_filt + n_mu;
  float* mu_pred = sigma_filt + n_sig;
  float* sigma_pred = mu_pred + n_mu;
  float* lat_means = sigma_pred + n_sig;
  float* lat_vars = lat_means + n_lm;
  float* S_out = lat_vars + n_lv;

  hipLaunchKernelGGL(fill_latents, dim3(2048), dim3(256), 0, stream, lat_means,
                     lat_vars);
  hipLaunchKernelGGL(kalman_fwd, dim3(BB), dim3(32), 0, stream, obs, A, C, D,
                     mu_filt, sigma_filt, mu_pred, sigma_pred, lat_means,
                     S_out);
}